// QNetwork_85272280694876
// MI455X (gfx1250) — compile-verified
//
#include <hip/hip_runtime.h>

// Problem constants (match reference)
static constexpr int B_ = 4;
static constexpr int N_ = 100000;
static constexpr long long E_ = 1600000;
static constexpr int D_ = 64;
static constexpr int T_ = 4;

typedef __attribute__((ext_vector_type(2))) float v2f;
typedef __attribute__((ext_vector_type(8))) float v8f;

// ---------------- utility: zero-fill ----------------
__global__ void k_zero(float* __restrict__ p, int n) {
    int i = blockIdx.x * blockDim.x + threadIdx.x;
    if (i < n) p[i] = 0.0f;
}

// ---------------- weighted in-degree: sdeg[u] = sum_{dst=u} w_e ----------------
__global__ void k_sdeg(const int* __restrict__ edst, const float* __restrict__ ew,
                       float* __restrict__ sdeg, int nE) {
    int e = blockIdx.x * blockDim.x + threadIdx.x;
    if (e < nE) atomicAdd(&sdeg[edst[e]], ew[e]);
}

// ---------------- c0[d] = sum_k theta3[d,k] * relu(theta4[k]) ----------------
__global__ void k_c0(const float* __restrict__ th3, const float* __restrict__ th4,
                     float* __restrict__ c0) {
    int d = threadIdx.x;  // 64 threads
    float s = 0.0f;
    for (int k = 0; k < D_; ++k) s += th3[d * D_ + k] * fmaxf(th4[k], 0.0f);
    c0[d] = s;
}

// ---------------- mu1 = relu(base), float4-vectorized ----------------
__global__ __launch_bounds__(256)
void k_mu_init(const float* __restrict__ xvs_b, const float* __restrict__ sdeg,
               const float* __restrict__ th1, const float* __restrict__ c0,
               float* __restrict__ mu) {
    int i = blockIdx.x * blockDim.x + threadIdx.x;  // over N*D/4
    if (i >= N_ * D_ / 4) return;
    int u  = i >> 4;          // node = (4i)/64
    int d0 = (i & 15) * 4;    // column base
    float xv = xvs_b[u], sd = sdeg[u];
    float4 t1 = *reinterpret_cast<const float4*>(th1 + d0);
    float4 cc = *reinterpret_cast<const float4*>(c0 + d0);
    float4 r;
    r.x = fmaxf(xv * t1.x + sd * cc.x, 0.0f);
    r.y = fmaxf(xv * t1.y + sd * cc.y, 0.0f);
    r.z = fmaxf(xv * t1.z + sd * cc.z, 0.0f);
    r.w = fmaxf(xv * t1.w + sd * cc.w, 0.0f);
    *reinterpret_cast<float4*>(mu + (size_t)i * 4) = r;
}

// ---------------- edge scatter: agg[dst] += mu[src]  (one wave per edge) ----------------
__global__ __launch_bounds__(256)
void k_scatter(const int* __restrict__ esrc, const int* __restrict__ edst,
               const float* __restrict__ mu, float* __restrict__ agg, int nE) {
    int gwave = (blockIdx.x * 256 + threadIdx.x) >> 5;
    int lane  = threadIdx.x & 31;
    if (gwave >= nE) return;
    // keep the HBM index streams ahead of the L2-atomic critical path
    if (lane == 0) {
        int pf = gwave + 4096;
        if (pf < nE) {
            __builtin_prefetch(esrc + pf, 0, 1);
            __builtin_prefetch(edst + pf, 0, 1);
        }
    }
    int s = esrc[gwave];
    int t = edst[gwave];
    const float2 v = *reinterpret_cast<const float2*>(mu + (size_t)s * D_ + 2 * lane);
    float* a = agg + (size_t)t * D_ + 2 * lane;
    atomicAdd(a,     v.x);
    atomicAdd(a + 1, v.y);
}

// ---------------- mu = relu(base + agg @ theta2^T) via V_WMMA_F32_16X16X4_F32 ----------------
// Block: 256 threads = 8 waves; covers 32 nodes x 64 out dims (one 16x16 tile per wave).
// K=64 accumulated as two independent 8-chunk WMMA chains (even/odd k-blocks) for ILP.
// Also writes zeros back to the consumed agg tile so no separate zero pass is needed.
__global__ __launch_bounds__(256)
void k_step_gemm(float* __restrict__ agg, const float* __restrict__ theta2,
                 const float* __restrict__ xvs_b, const float* __restrict__ sdeg,
                 const float* __restrict__ th1, const float* __restrict__ c0,
                 float* __restrict__ mu) {
    __shared__ float sB[D_ * D_];   // theta2 (row-major [d][k]), 16 KB
    __shared__ float sA[32 * D_];   // agg tile: 32 nodes x 64 k, 8 KB
    __shared__ float sxv[32];
    __shared__ float sdg[32];

    const int tid = threadIdx.x;
    const int nodeBase = blockIdx.x * 32;

    for (int i = tid; i < D_ * D_; i += 256) sB[i] = theta2[i];
    for (int i = tid; i < 32 * D_; i += 256) {
        float v = agg[(size_t)nodeBase * D_ + i];
        sA[i] = v;
        agg[(size_t)nodeBase * D_ + i] = 0.0f;   // pre-zero for the next scatter pass
    }
    if (tid < 32) {
        sxv[tid] = xvs_b[nodeBase + tid];
        sdg[tid] = sdeg[nodeBase + tid];
    }
    __syncthreads();

    const int wave = tid >> 5;
    const int lane = tid & 31;
    const int nt = wave >> 2;          // node tile 0..1
    const int dt = wave & 3;           // d tile 0..3
    const int m  = lane & 15;          // A row (M) / B,C column (N)
    const int halfk = (lane < 16) ? 0 : 2;

    // ISA layouts (CDNA5 ISA):
    //  A 16x4 f32: lane L holds A[M=L%16][K=halfk], A[M][K=halfk+1] in 2 VGPRs
    //  B 4x16 f32: lane L holds B[K=halfk][N=L%16], B[halfk+1][N]
    //  C/D 16x16:  lane L, reg r -> (M = r + (L<16?0:8), N = L%16)
    const float* aRow = sA + (nt * 16 + m) * D_;   // agg row of node (nodeBase+nt*16+m)
    const float* bRow = sB + (dt * 16 + m) * D_;   // theta2 row d = dt*16+m

    v8f acc0 = {};
    v8f acc1 = {};
    #pragma unroll
    for (int kk = 0; kk < D_; kk += 8) {
        v2f a0 = *reinterpret_cast<const v2f*>(aRow + kk + halfk);
        v2f b0 = *reinterpret_cast<const v2f*>(bRow + kk + halfk);
        v2f a1 = *reinterpret_cast<const v2f*>(aRow + kk + 4 + halfk);
        v2f b1 = *reinterpret_cast<const v2f*>(bRow + kk + 4 + halfk);
        acc0 = __builtin_amdgcn_wmma_f32_16x16x4_f32(false, a0, false, b0,
                                                     (short)0, acc0, false, false);
        acc1 = __builtin_amdgcn_wmma_f32_16x16x4_f32(false, a1, false, b1,
                                                     (short)0, acc1, false, false);
    }

    // Epilogue: combine chains, add base, relu, store.
    const int dcol = dt * 16 + m;
    const float t1 = th1[dcol];
    const float cc = c0[dcol];
    const int rbase = (lane < 16) ? 0 : 8;
    #pragma unroll
    for (int r = 0; r < 8; ++r) {
        const int ul = nt * 16 + rbase + r;        // local node index in [0,32)
        float val = (acc0[r] + acc1[r]) + sxv[ul] * t1 + sdg[ul] * cc;
        mu[(size_t)(nodeBase + ul) * D_ + dcol] = fmaxf(val, 0.0f);
    }
}

// ---------------- pooled[b,d] = sum_u mu[u,d] ----------------
__global__ __launch_bounds__(256)
void k_reduce(const float* __restrict__ mu, float* __restrict__ pooled_b) {
    int tid = threadIdx.x;
    int d = tid & 63, g = tid >> 6;  // 4 row-groups per block
    float s = 0.0f;
    for (int u = blockIdx.x * 4 + g; u < N_; u += gridDim.x * 4)
        s += mu[(size_t)u * D_ + d];
    atomicAdd(&pooled_b[d], s);
}

// ---------------- save mu[vs[b], :] ----------------
__global__ void k_saverow(const float* __restrict__ mu, const int* __restrict__ vs,
                          int b, float* __restrict__ localrow) {
    int d = threadIdx.x;  // 64
    localrow[b * D_ + d] = mu[(size_t)vs[b] * D_ + d];
}

// ---------------- readout: q[b] = relu([pooled@th6^T ; local@th7^T]) . th5 ----------------
__global__ void k_final(const float* __restrict__ pooled, const float* __restrict__ localrow,
                        const float* __restrict__ th5, const float* __restrict__ th6,
                        const float* __restrict__ th7, float* __restrict__ out) {
    __shared__ float red[B_ * D_];
    int tid = threadIdx.x;      // 256 = B*D
    int b = tid >> 6, d = tid & 63;
    float p2 = 0.0f, lc = 0.0f;
    for (int k = 0; k < D_; ++k) {
        p2 += th6[d * D_ + k] * pooled[b * D_ + k];
        lc += th7[d * D_ + k] * localrow[b * D_ + k];
    }
    red[tid] = fmaxf(p2, 0.0f) * th5[d] + fmaxf(lc, 0.0f) * th5[D_ + d];
    __syncthreads();
    if (d == 0) {
        float q = 0.0f;
        for (int k = 0; k < D_; ++k) q += red[b * D_ + k];
        out[b] = q;
    }
}

extern "C" void kernel_launch(void* const* d_in, const int* in_sizes, int n_in,
                              void* d_out, int out_size, void* d_ws, size_t ws_size,
                              hipStream_t stream) {
    const float* xvs  = (const float*)d_in[0];  // [B,N]
    const int*   vs   = (const int*)d_in[1];    // [B]
    const int*   esrc = (const int*)d_in[2];    // [E]
    const int*   edst = (const int*)d_in[3];    // [E]
    const float* ew   = (const float*)d_in[4];  // [E]
    const float* th1  = (const float*)d_in[5];  // [D]
    const float* th2  = (const float*)d_in[6];  // [D,D]
    const float* th3  = (const float*)d_in[7];  // [D,D]
    const float* th4  = (const float*)d_in[8];  // [D]
    const float* th5  = (const float*)d_in[9];  // [2D]
    const float* th6  = (const float*)d_in[10]; // [D,D]
    const float* th7  = (const float*)d_in[11]; // [D,D]
    float* out = (float*)d_out;

    const int nE = (int)E_;

    // Workspace layout (floats). Batch-sequential so mu+agg (51 MB) stays L2-resident.
    float* ws       = (float*)d_ws;
    float* sdeg     = ws;                          // N
    float* c0       = sdeg + N_;                   // D
    float* pooled   = c0 + D_;                     // B*D
    float* localrow = pooled + B_ * D_;            // B*D
    float* mu       = localrow + B_ * D_;          // N*D
    float* agg      = mu + (size_t)N_ * D_;        // N*D

    const int ND = N_ * D_;

    // Precompute: weighted in-degree + collapsed weight-message vector.
    k_zero<<<(N_ + 255) / 256, 256, 0, stream>>>(sdeg, N_);
    k_sdeg<<<(nE + 255) / 256, 256, 0, stream>>>(edst, ew, sdeg, nE);
    k_c0<<<1, 64, 0, stream>>>(th3, th4, c0);
    k_zero<<<1, 256, 0, stream>>>(pooled, B_ * D_);
    // agg zeroed once here; k_step_gemm re-zeroes the tiles it consumes thereafter.
    k_zero<<<(ND + 255) / 256, 256, 0, stream>>>(agg, ND);

    const int scatterBlocks = (nE * 32) / 256;     // one wave per edge

    for (int b = 0; b < B_; ++b) {
        const float* xvs_b = xvs + (size_t)b * N_;
        // t = 1: mu = relu(base) (agg of mu0=0 vanishes)
        k_mu_init<<<(ND / 4 + 255) / 256, 256, 0, stream>>>(xvs_b, sdeg, th1, c0, mu);
        // t = 2..T
        for (int t = 1; t < T_; ++t) {
            k_scatter<<<scatterBlocks, 256, 0, stream>>>(esrc, edst, mu, agg, nE);
            k_step_gemm<<<N_ / 32, 256, 0, stream>>>(agg, th2, xvs_b, sdeg, th1, c0, mu);
        }
        k_reduce<<<250, 256, 0, stream>>>(mu, pooled + b * D_);
        k_saverow<<<1, 64, 0, stream>>>(mu, vs, b, localrow);
    }
    k_final<<<1, 256, 0, stream>>>(pooled, localrow, th5, th6, th7, out);
}